// YOLOLoss_21251498181369
// MI455X (gfx1250) — compile-verified
//
#include <hip/hip_runtime.h>
#include <hip/hip_bf16.h>
#include <stdint.h>

// ---------------- problem constants (from setup_inputs) ----------------
#define BS   32
#define TT   50
#define NC   80
#define HH   76
#define WW   76
#define HW   (HH * WW)          // 5776
#define NA   (5 + NC)           // 85
#define SIGMA 8.0f

#define TILE 256                // cells per tile (== blockDim.x of main kernel)
#define BPB  8                  // blocks per batch image
#define NTILES ((HW + TILE - 1) / TILE)   // 23 (last tile = 144 cells)

// ---------------- CDNA5 feature probes ----------------
typedef unsigned int u32x4 __attribute__((ext_vector_type(4)));
typedef int          i32x4 __attribute__((ext_vector_type(4)));
typedef int          i32x8 __attribute__((ext_vector_type(8)));
typedef float        v2f   __attribute__((ext_vector_type(2)));
typedef float        v8f   __attribute__((ext_vector_type(8)));

#if defined(__has_builtin)
#if __has_builtin(__builtin_amdgcn_tensor_load_to_lds) && __has_builtin(__builtin_amdgcn_s_wait_tensorcnt)
#define HAVE_TDM 1
#endif
#if __has_builtin(__builtin_amdgcn_wmma_f32_16x16x4_f32)
#define HAVE_WMMA_F32 1
#endif
#endif
#ifndef HAVE_TDM
#define HAVE_TDM 0
#endif
#ifndef HAVE_WMMA_F32
#define HAVE_WMMA_F32 0
#endif

#if HAVE_TDM
#pragma message("CDNA5 probe: tensor_load_to_lds builtin AVAILABLE (TDM path)")
#else
#pragma message("CDNA5 probe: tensor_load_to_lds builtin MISSING (staged-load fallback)")
#endif
#if HAVE_WMMA_F32
#pragma message("CDNA5 probe: wmma_f32_16x16x4_f32 builtin AVAILABLE (matrix-core reduction)")
#else
#pragma message("CDNA5 probe: wmma_f32_16x16x4_f32 builtin MISSING (LDS-tree reduction)")
#endif

#if HAVE_TDM
// 2-D tile DMA: dim0 = tdim contiguous f32 cells, dim1 = 5 channels,
// channel stride = HW elements. D# packed per CDNA5 ISA ch.8 (§8.3/8.4).
__device__ __forceinline__ void tdm_load_tile(const float* gsrc, unsigned lds_off, int tdim) {
    unsigned long long ga = (unsigned long long)(uintptr_t)gsrc;
    u32x4 g0;
    i32x8 g1;
    i32x4 g2 = {0, 0, 0, 0}, g3 = {0, 0, 0, 0};
    // group0: [1:0]=count=1, [63:32]=lds_addr, [120:64]=global_addr, [127:126]=type=2
    g0.x = 1u;
    g0.y = lds_off;
    g0.z = (unsigned)(ga & 0xFFFFFFFFull);
    g0.w = (unsigned)((ga >> 32) & 0x1FFFFFFull) | 0x80000000u;
    // group1
    const unsigned td0 = (unsigned)HW;   // tensor_dim0
    const unsigned td1 = (unsigned)NA;   // tensor_dim1
    const unsigned tl0 = (unsigned)tdim; // tile_dim0 (<= 256)
    const unsigned tl1 = 5u;             // tile_dim1 (channels 0..4)
    const unsigned long long s0 = (unsigned long long)HW; // tensor_dim0_stride
    g1[0] = (int)(2u << 16);                                   // data_size = 4B
    g1[1] = (int)((td0 & 0xFFFFu) << 16);                      // td0[15:0] @ bits 63:48
    g1[2] = (int)((td0 >> 16) | ((td1 & 0xFFFFu) << 16));      // td0[31:16] | td1[15:0]
    g1[3] = (int)((td1 >> 16) | (tl0 << 16));                  // td1[31:16] | tile_dim0
    g1[4] = (int)(tl1);                                        // tile_dim1 | tile_dim2=0
    g1[5] = (int)(s0 & 0xFFFFFFFFull);                         // stride0[31:0]
    g1[6] = (int)((s0 >> 32) & 0xFFFFu);                       // stride0[47:32] | stride1 lo
    g1[7] = 0;
#if defined(__clang_major__) && (__clang_major__ >= 23)
    i32x8 g4 = {0, 0, 0, 0, 0, 0, 0, 0};
    __builtin_amdgcn_tensor_load_to_lds(g0, g1, g2, g3, g4, 0);
#else
    __builtin_amdgcn_tensor_load_to_lds(g0, g1, g2, g3, 0);
#endif
}
#endif // HAVE_TDM

// ---------------- math helpers (match reference semantics) ----------------
__device__ __forceinline__ float sigf(float v) { return 1.0f / (1.0f + expf(-v)); }
__device__ __forceinline__ float clogf_(float p) {
    p = fminf(fmaxf(p, 1e-12f), 1.0f);
    float l = logf(p);
    return fminf(fmaxf(l, -100.0f), 0.0f);
}
__device__ __forceinline__ float bcef(float p, float t) {
    return -(t * clogf_(p) + (1.0f - t) * clogf_(1.0f - p));
}
__device__ __forceinline__ float sl1f(float a, float b) {
    float d = fabsf(a - b);
    return d < 1.0f ? 0.5f * d * d : d - 0.5f;
}

// ---------------- kernel 1: target assignment (per batch image) ----------------
// tinfo layout per slot (12 f32): gx gy gw gh cell tx ty tw th scale cls pad
__global__ void yolo_targets(const float* __restrict__ tgts,
                             int* __restrict__ nEff,
                             float* __restrict__ tinfo) {
    __shared__ int s_cell[TT];
    __shared__ unsigned char s_valid[TT];
    __shared__ unsigned char s_eff[TT];
    const int b = blockIdx.x, t = threadIdx.x;
    float v0 = 0, v1 = 0, v2 = 0, v3 = 0, v4 = 0;
    int cell = 0, valid = 0;
    if (t < TT) {
        const float* p = tgts + (size_t)(b * TT + t) * 5;
        v0 = p[0]; v1 = p[1]; v2 = p[2]; v3 = p[3]; v4 = p[4];
        valid = ((v0 + v1 + v2 + v3 + v4) != 0.0f);
        float gx = v1 * (float)WW, gy = v2 * (float)HH;
        int gi = min(max((int)gx, 0), WW - 1);
        int gj = min(max((int)gy, 0), HH - 1);
        cell = gj * WW + gi;
        s_cell[t] = cell;
        s_valid[t] = (unsigned char)valid;
    }
    __syncthreads();
    int eff = 0;
    if (t < TT && valid) {
        int dup = 0;
        for (int u = 0; u < t; ++u) dup |= (s_valid[u] && (s_cell[u] == cell));
        eff = !dup;
    }
    if (t < TT) s_eff[t] = (unsigned char)eff;
    __syncthreads();
    if (eff) {
        int slot = 0;
        for (int u = 0; u < t; ++u) slot += s_eff[u];   // deterministic compaction
        float gx = v1 * (float)WW, gy = v2 * (float)HH;
        float gw = v3 * (float)WW, gh = v4 * (float)HH;
        int gi = cell % WW, gj = cell / WW;
        float* o = tinfo + (size_t)(b * TT + slot) * 12;
        o[0] = gx; o[1] = gy; o[2] = gw; o[3] = gh;
        o[4] = (float)cell;
        o[5] = gx - (float)gi;                 // tx
        o[6] = gy - (float)gj;                 // ty
        o[7] = logf(gw * (1.0f / SIGMA) + 1e-16f);  // tw
        o[8] = logf(gh * (1.0f / SIGMA) + 1e-16f);  // th
        o[9] = 2.0f - v3 * v4;                 // scale
        o[10] = (float)min(max((int)v0, 0), NC - 1);
        o[11] = 0.0f;
    }
    __syncthreads();
    if (t == 0) {
        int n = 0;
        for (int u = 0; u < TT; ++u) n += s_eff[u];
        nEff[b] = n;
    }
}

// ---------------- kernel 2: main per-cell loss ----------------
__global__ __launch_bounds__(TILE) void yolo_main(const float* __restrict__ inp,
                                                  const float* __restrict__ tinfo,
                                                  const int* __restrict__ nEff,
                                                  float* __restrict__ partial) {
    const int b = blockIdx.y, bx = blockIdx.x, tid = threadIdx.x;
    __shared__ float sT[TT * 12];
    __shared__ float sBuf[2][5 * TILE];
    __shared__ float sRed[128 * 16];     // padded [128 rows x 16 cols] partials matrix
    __shared__ float sCol[16];

    const int nt = nEff[b];
    for (int i = tid; i < nt * 12; i += TILE) sT[i] = tinfo[(size_t)b * TT * 12 + i];

    int tiles[(NTILES + BPB - 1) / BPB];
    int ntile = 0;
    for (int t = bx; t < NTILES; t += BPB) tiles[ntile++] = t;

    const float* gbase = inp + (size_t)b * NA * HW;

#if HAVE_TDM
    if (tid < 32 && ntile > 0) {
        const int t0 = tiles[0];
        const int td = min(TILE, HW - t0 * TILE);
        tdm_load_tile(gbase + t0 * TILE, (unsigned)(uintptr_t)&sBuf[0][0], td);
    }
#endif

    float v0 = 0, v1 = 0, v2 = 0, v3 = 0, v4 = 0, v5 = 0; // lx ly lw lh lconf lcls

    for (int k = 0; k < ntile; ++k) {
#if HAVE_TDM
        if (tid < 32) {
            if (k + 1 < ntile) {  // double-buffer: issue next tile, wait for current
                const int tn = tiles[k + 1];
                const int tdn = min(TILE, HW - tn * TILE);
                tdm_load_tile(gbase + tn * TILE, (unsigned)(uintptr_t)&sBuf[(k + 1) & 1][0], tdn);
                __builtin_amdgcn_s_wait_tensorcnt(1);
            } else {
                __builtin_amdgcn_s_wait_tensorcnt(0);
            }
        }
#else
        { // fallback: cooperative staged load of tile k
            const int tc = tiles[k];
            const int tdim = min(TILE, HW - tc * TILE);
            const float* src = gbase + tc * TILE;
            for (int i = tid; i < 5 * tdim; i += TILE) {
                const int ch = i / tdim, xx = i - ch * tdim;
                sBuf[k & 1][i] = src[(size_t)ch * HW + xx];
            }
        }
#endif
        __syncthreads();

        const int tc = tiles[k];
        const int tdim = min(TILE, HW - tc * TILE);
        const int base = tc * TILE;
        const float* buf = sBuf[k & 1];
        if (tid < tdim) {
            const int cell = base + tid;
            const int ci = cell % WW, cj = cell / WW;
            const float x = sigf(buf[tid]);
            const float y = sigf(buf[tdim + tid]);
            const float w = buf[2 * tdim + tid];
            const float h = buf[3 * tdim + tid];
            const float conf = sigf(buf[4 * tdim + tid]);
            const float pxc = x + (float)ci, pyc = y + (float)cj;
            const float pw = expf(w) * SIGMA, ph = expf(h) * SIGMA;
            const float parea = pw * ph;

            bool ignore = false;
            int match = -1;
            for (int t = 0; t < nt; ++t) {
                const float* T = &sT[t * 12];
                const float gx = T[0], gy = T[1], gw = T[2], gh = T[3];
                float iw = fminf(gx + gw * 0.5f, pxc + pw * 0.5f) - fmaxf(gx - gw * 0.5f, pxc - pw * 0.5f);
                float ih = fminf(gy + gh * 0.5f, pyc + ph * 0.5f) - fmaxf(gy - gh * 0.5f, pyc - ph * 0.5f);
                iw = fmaxf(iw, 0.0f); ih = fmaxf(ih, 0.0f);
                const float inter = iw * ih;
                const float iou = inter / (gw * gh + parea - inter + 1e-16f);
                ignore = ignore || (iou >= 0.5f);
                if ((int)T[4] == cell) match = t;
            }
            if (match >= 0) {           // mask==1 cell (unique assigned target)
                const float* T = &sT[match * 12];
                const float sc = T[9];
                v0 += sc * bcef(x, T[5]);
                v1 += sc * bcef(y, T[6]);
                v2 += sc * sl1f(w, T[7]);
                v3 += sc * sl1f(h, T[8]);
                v4 += -clogf_(conf);    // BCE(conf,1); noobj term = 0 here
                const int cls = (int)T[10];
                const float* pc = gbase + (size_t)5 * HW + cell;
                for (int c = 0; c < NC; ++c) {
                    const float p = sigf(pc[(size_t)c * HW]);
                    v5 += bcef(p, (c == cls) ? 1.0f : 0.0f);
                }
            } else if (!ignore) {       // noobj cell
                v4 += -0.5f * clogf_(1.0f - conf);
            }                           // ignored non-mask cell contributes 0
        }
        __syncthreads();
    }

    // ---- block reduction of the 6 partial sums ----
    // Layout: thread t -> row t/2, column offset (t&1)*8; cols 6,7 zero-padded.
    {
        const int base = (tid >> 1) * 16 + (tid & 1) * 8;
        sRed[base + 0] = v0; sRed[base + 1] = v1; sRed[base + 2] = v2;
        sRed[base + 3] = v3; sRed[base + 4] = v4; sRed[base + 5] = v5;
        sRed[base + 6] = 0.0f; sRed[base + 7] = 0.0f;
    }
    __syncthreads();

#if HAVE_WMMA_F32
    // Matrix-core column-sum: C <- Ones(16x4) x B(4x16) + C over 32 chunks of 4 rows.
    // B VGPR layout (ISA 7.12.2): elem0 = row {0|2}, elem1 = row {1|3}, lanes 0-15 / 16-31.
    if (tid < 32) {                      // wave 0, EXEC all ones
        const int col = tid & 15;
        const int rowOff = (tid < 16) ? 0 : 2;
        const v2f onesA = {1.0f, 1.0f};
        v8f acc8 = {0.0f, 0.0f, 0.0f, 0.0f, 0.0f, 0.0f, 0.0f, 0.0f};
        for (int i = 0; i < 32; ++i) {
            v2f bv;
            bv.x = sRed[(4 * i + rowOff) * 16 + col];
            bv.y = sRed[(4 * i + rowOff + 1) * 16 + col];
            acc8 = __builtin_amdgcn_wmma_f32_16x16x4_f32(
                false, onesA, false, bv, (short)0, acc8, false, false);
        }
        if (tid < 16) sCol[tid] = acc8[0];   // D[0][n] lives in C-VGPR0, lanes 0-15
    }
#else
    for (int s = 64; s > 0; s >>= 1) {       // tree over rows of the [128][16] matrix
        for (int idx = tid; idx < s * 16; idx += TILE)
            sRed[idx] += sRed[idx + s * 16];
        __syncthreads();
    }
    if (tid < 16) sCol[tid] = sRed[tid];
#endif
    __syncthreads();
    if (tid == 0) {
        float* o = partial + (size_t)(b * BPB + bx) * 6;
        for (int c = 0; c < 6; ++c) o[c] = sCol[c] + sCol[c + 8];  // even + odd threads
    }
}

// ---------------- kernel 3: final deterministic reduction + output ----------------
__global__ void yolo_final(const float* __restrict__ partial,
                           const int* __restrict__ nEff,
                           float* __restrict__ out) {
    __shared__ float red[BPB * BS][6];
    const int tid = threadIdx.x;   // 256 threads == 256 partials
    for (int c = 0; c < 6; ++c) red[tid][c] = partial[(size_t)tid * 6 + c];
    __syncthreads();
    for (int s = (BPB * BS) / 2; s > 0; s >>= 1) {
        if (tid < s)
            for (int c = 0; c < 6; ++c) red[tid][c] += red[tid + s][c];
        __syncthreads();
    }
    if (tid == 0) {
        int n = 0;
        for (int bb = 0; bb < BS; ++bb) n += nEff[bb];
        const float fn = (float)n;
        const float lx = red[0][0] / fn, ly = red[0][1] / fn;
        const float lw = red[0][2] / fn, lh = red[0][3] / fn;
        const float lc = red[0][4] / fn;
        const float lt = (n > 0) ? red[0][5] / fn : 0.0f;   // sum(tcls)>0 gate == n_obj>0
        out[0] = lx + ly + lw + lh + lc + lt;
        out[1] = lx; out[2] = ly; out[3] = lw; out[4] = lh; out[5] = lc; out[6] = lt;
    }
}

// ---------------- host launcher ----------------
extern "C" void kernel_launch(void* const* d_in, const int* in_sizes, int n_in,
                              void* d_out, int out_size, void* d_ws, size_t ws_size,
                              hipStream_t stream) {
    (void)in_sizes; (void)n_in; (void)out_size; (void)ws_size;
    const float* inp  = (const float*)d_in[0];
    const float* tgts = (const float*)d_in[1];
    float* out = (float*)d_out;

    char* ws = (char*)d_ws;
    int*   nEff    = (int*)ws;                                   // 32 ints
    float* tinfo   = (float*)(ws + 128);                         // 32*50*12 f32
    float* partial = (float*)(ws + 128 + (size_t)BS * TT * 12 * 4); // 256*6 f32

    yolo_targets<<<BS, 64, 0, stream>>>(tgts, nEff, tinfo);
    dim3 grid(BPB, BS);
    yolo_main<<<grid, TILE, 0, stream>>>(inp, tinfo, nEff, partial);
    yolo_final<<<1, BPB * BS, 0, stream>>>(partial, nEff, out);
}